// TimeEmbedding_33801392620091
// MI455X (gfx1250) — compile-verified
//
#include <hip/hip_runtime.h>

// TimeEmbedding: out[b,d] = d even ? sin(t[b]*invf(d)) : cos(t[b]*invf(d))
// invf(d) = 10000^(-2d/512). Implemented as sin(2*pi*f) with
//   f = t[b] * (invf(d)/2pi) + (d odd ? 0.25 : 0)
// The affine map f = A x B + C is computed on the matrix pipe with
// V_WMMA_F32_16X16X4_F32 (rank-1 outer product, K0 only), leaving the
// vector pipes to do only v_sin_f32 + non-temporal streaming stores.

typedef float v2f __attribute__((ext_vector_type(2)));
typedef float v8f __attribute__((ext_vector_type(8)));

#define DMODEL 512

__global__ __launch_bounds__(256) void TimeEmbedding_wmma_kernel(
    const float* __restrict__ t, float* __restrict__ out) {
  const int lane = threadIdx.x & 31;
  const int wave = threadIdx.x >> 5;
  const int col  = lane & 15;   // N within 16x16 tile
  const int half = lane >> 4;   // 0 -> rows M=r, 1 -> rows M=r+8 (also K-group in A/B)
  const int rowBase = blockIdx.x * 128 + wave * 16;

  // ---- A matrix (16x4 f32, only K=0 column non-zero) ----
  // layout: lanes 0-15 hold K0/K1 in v0/v1; lanes 16-31 hold K2/K3.
  const float tval = t[rowBase + col];   // lanes 16-31 re-read same 16 values (broadcast)
  v2f a;
  a.x = (half == 0) ? tval : 0.0f;       // K0 rows = t, K2 = 0
  a.y = 0.0f;                            // K1 / K3 = 0

  // ---- B matrix (4x16 f32, only row K=0 non-zero) ----
  // b0(col) = invf(col)/(2*pi) = exp2(-col * log2(10000)/256) * (1/2pi)
  const float SCOL   = 0.051905126482615034f;  // log2(10000)/256
  const float K16    = 0.5623413251903491f;    // 10000^(-1/16): step for colBase += 16
  const float INV2PI = 0.15915494309189535f;
  float b0 = __builtin_amdgcn_exp2f(-SCOL * (float)col) * INV2PI;
  b0 = (half == 0) ? b0 : 0.0f;                // K2/K3 rows = 0
  v2f bvec;
  bvec.y = 0.0f;                               // K1 / K3 row = 0

  // ---- C matrix: per-column phase bias (0 / 0.25 revolutions) ----
  const float cb = (col & 1) ? 0.25f : 0.0f;
  v8f cvec;
#pragma unroll
  for (int r = 0; r < 8; ++r) cvec[r] = cb;

  // lane's output base: row (rowBase + 8*half + r), column col
  float* __restrict__ p =
      out + (size_t)(rowBase + 8 * half) * DMODEL + col;

#pragma unroll 4
  for (int cb0 = 0; cb0 < DMODEL; cb0 += 16) {
    bvec.x = b0;
    // D = t (x) invf' + phase  -> 16x16 tile of sine arguments (revolutions)
    v8f d = __builtin_amdgcn_wmma_f32_16x16x4_f32(
        /*neg_a=*/false, a, /*neg_b=*/false, bvec,
        /*c_mod=*/(short)0, cvec, /*reuse_a=*/false, /*reuse_b=*/false);

    float* __restrict__ q = p + cb0;
#pragma unroll
    for (int r = 0; r < 8; ++r) {
      // v_sin_f32 consumes revolutions; stores stream with TH=NT
      __builtin_nontemporal_store(__builtin_amdgcn_sinf(d[r]), q + r * DMODEL);
    }
    b0 *= K16;  // advance inv_freq by 16 columns
  }
}

extern "C" void kernel_launch(void* const* d_in, const int* in_sizes, int n_in,
                              void* d_out, int out_size, void* d_ws, size_t ws_size,
                              hipStream_t stream) {
  const float* t = (const float*)d_in[0];
  float* out = (float*)d_out;
  const int n = in_sizes[0];        // 524288 rows (multiple of 128 by construction)
  const int blocks = n / 128;       // 8 waves/block * 16 rows/wave
  TimeEmbedding_wmma_kernel<<<blocks, 256, 0, stream>>>(t, out);
}